// RNN_30279519437218
// MI455X (gfx1250) — compile-verified
//
#include <hip/hip_runtime.h>
#include <cmath>
#include <stdint.h>

// Problem constants (from reference)
#define SEQ_LEN   4096
#define BATCH_N   32
#define IN_SZ     471
#define HID       5
#define NCLS      5
#define ROWS      (BATCH_N * SEQ_LEN)   // 131072
#define TILE_F4   ((16 * IN_SZ) / 4)    // 1884 16-byte chunks per 16-row tile

typedef float v2f __attribute__((ext_vector_type(2)));
typedef float v8f __attribute__((ext_vector_type(8)));

// ---------------------------------------------------------------------------
// Stage 1: xw[row][h] = sum_k x[row][k] * W_ih[h][k] + (b_ih[h] + b_hh[h])
// One wave (32 threads) per 16-row tile of x.
//  - x tile staged global->LDS with ASYNC b128 copies (ASYNCcnt-tracked,
//    no VGPR bounce); tile is contiguous in memory so the copy is 1:1.
//  - f32 WMMA 16x16x4: 117 uniform K-steps + 1 masked tail step (K=471).
// ---------------------------------------------------------------------------
__global__ __launch_bounds__(32) void rnn_proj_wmma(
    const float* __restrict__ x, const float* __restrict__ W_ih,
    const float* __restrict__ b_ih, const float* __restrict__ b_hh,
    float* __restrict__ xw)
{
  // +2 floats of slack so masked-away tail reads stay in bounds.
  __shared__ __attribute__((aligned(16))) float sA[16 * IN_SZ + 2];  // 30152 B
  __shared__ __attribute__((aligned(16))) float sB[HID * IN_SZ + 2]; //  9428 B

  const int  lane = threadIdx.x;
  const long row0 = (long)blockIdx.x * 16;

  // ---- async global->LDS staging of the 16-row x tile (flat layout) ----
  {
    const uint64_t gbase   = (uint64_t)(uintptr_t)(x + row0 * IN_SZ);
    const uint32_t ldsbase = (uint32_t)(uintptr_t)sA;  // low 32 bits = LDS offset
    for (int i = lane; i < TILE_F4; i += 32) {
      uint64_t ga = gbase + (uint64_t)i * 16u;
      uint32_t la = ldsbase + (uint32_t)i * 16u;
      asm volatile("global_load_async_to_lds_b128 %0, %1, off"
                   :: "v"(la), "v"(ga) : "memory");
    }
  }

  // ---- stage W_ih (5 x 471) into LDS via normal path (tiny, cached) ----
  for (int i = lane; i < HID * IN_SZ; i += 32) sB[i] = W_ih[i];

  asm volatile("s_wait_asynccnt 0x0" ::: "memory");  // async tile landed
  __syncthreads();                                   // + dscnt for sB stores

  // Fragment addressing per ISA 32-bit 16x4 A layout:
  //   lanes 0-15 : K = k+0, k+1      lanes 16-31 : K = k+2, k+3
  const int   r     = lane & 15;           // A row-in-tile == B/C/D column
  const int   koff  = (lane >> 4) << 1;
  const float bmask = (r < HID) ? 1.0f : 0.0f;       // N >= 5 columns zero
  const float* pa = sA + r * IN_SZ + koff;
  const float* pb = sB + ((r < HID) ? r : (HID - 1)) * IN_SZ + koff;

  v8f acc = {};
#pragma unroll 2
  for (int kc = 0; kc < IN_SZ / 4; ++kc) {           // 117 uniform K=4 steps
    v2f a = { pa[4 * kc], pa[4 * kc + 1] };
    v2f b = { pb[4 * kc] * bmask, pb[4 * kc + 1] * bmask };
    // (neg_a, A, neg_b, B, c_mod, C, reuse_a, reuse_b)
    acc = __builtin_amdgcn_wmma_f32_16x16x4_f32(false, a, false, b,
                                                (short)0, acc, false, false);
  }
  {
    // K tail: elements 468..470 valid; hi half-wave's second slot is K=471 -> 0
    const bool lo = (koff == 0);
    v2f a = { pa[468], lo ? pa[469] : 0.0f };
    v2f b = { pb[468] * bmask, lo ? pb[469] * bmask : 0.0f };
    acc = __builtin_amdgcn_wmma_f32_16x16x4_f32(false, a, false, b,
                                                (short)0, acc, false, false);
  }

  // C/D layout: VGPR j, lanes 0-15 -> M=j ; lanes 16-31 -> M=8+j ; N = lane&15
  if (r < HID) {
    const float cb    = b_ih[r] + b_hh[r];
    const int   mbase = (lane >> 4) * 8;
#pragma unroll
    for (int j = 0; j < 8; ++j) {
      long m = row0 + mbase + j;
      xw[m * HID + r] = acc[j] + cb;
    }
  }
}

// ---------------------------------------------------------------------------
// Stage 2: serial recurrence. One wave; lane b owns batch b's 4096-step chain.
// Hardware v_tanh_f32 (builtin if declared, else inline asm + TRANS-hazard nop).
// ---------------------------------------------------------------------------
__device__ __forceinline__ float fast_tanh(float v) {
#if __has_builtin(__builtin_amdgcn_tanhf)
  return __builtin_amdgcn_tanhf(v);
#elif __has_builtin(__builtin_amdgcn_tanh_f32)
  return __builtin_amdgcn_tanh_f32(v);
#else
  float r;
  asm volatile("v_tanh_f32 %0, %1\n\tv_nop" : "=v"(r) : "v"(v));
  return r;
#endif
}

__global__ __launch_bounds__(32) void rnn_recurrence(
    const float* __restrict__ xw, const float* __restrict__ W_hh,
    float* __restrict__ hs)
{
  const int b = threadIdx.x;     // batch index, 0..31
  float W[HID][HID];
#pragma unroll
  for (int j = 0; j < HID; ++j)
#pragma unroll
    for (int k = 0; k < HID; ++k) W[j][k] = W_hh[j * HID + k];

  float h[HID] = {0.f, 0.f, 0.f, 0.f, 0.f};
  const float* row  = xw + (long)b * SEQ_LEN * HID;
  float*       orow = hs + (long)b * SEQ_LEN * HID;

  for (int t = 0; t < SEQ_LEN; ++t) {
    __builtin_prefetch(row + 32 * HID, 0, 0);      // global_prefetch_b8 ahead
    float a[HID];
#pragma unroll
    for (int j = 0; j < HID; ++j) {
      float s = row[j];
#pragma unroll
      for (int k = 0; k < HID; ++k) s = fmaf(W[j][k], h[k], s);
      a[j] = s;
    }
#pragma unroll
    for (int j = 0; j < HID; ++j) {
      h[j] = fast_tanh(a[j]);
      orow[j] = h[j];
    }
    row  += HID;
    orow += HID;
  }
}

// ---------------------------------------------------------------------------
// Stage 3: out[row][c] = hs[row] . W_lin[c] + b_lin[c]   (tiny epilogue)
// ---------------------------------------------------------------------------
__global__ void rnn_linear(const float* __restrict__ hs,
                           const float* __restrict__ W_lin,
                           const float* __restrict__ b_lin,
                           float* __restrict__ out)
{
  long rr = (long)blockIdx.x * blockDim.x + threadIdx.x;
  if (rr >= ROWS) return;
  float h[HID];
#pragma unroll
  for (int k = 0; k < HID; ++k) h[k] = hs[rr * HID + k];
#pragma unroll
  for (int c = 0; c < NCLS; ++c) {
    float s = b_lin[c];
#pragma unroll
    for (int k = 0; k < HID; ++k) s = fmaf(W_lin[c * HID + k], h[k], s);
    out[rr * NCLS + c] = s;
  }
}

// ---------------------------------------------------------------------------
extern "C" void kernel_launch(void* const* d_in, const int* in_sizes, int n_in,
                              void* d_out, int out_size, void* d_ws, size_t ws_size,
                              hipStream_t stream) {
  (void)in_sizes; (void)n_in; (void)out_size; (void)ws_size;
  const float* x     = (const float*)d_in[0];
  const float* W_ih  = (const float*)d_in[1];
  const float* W_hh  = (const float*)d_in[2];
  const float* b_ih  = (const float*)d_in[3];
  const float* b_hh  = (const float*)d_in[4];
  const float* W_lin = (const float*)d_in[5];
  const float* b_lin = (const float*)d_in[6];
  float* out = (float*)d_out;

  float* xw = (float*)d_ws;                  // ROWS*HID f32  (~2.6 MB)
  float* hs = xw + (size_t)ROWS * HID;       // ROWS*HID f32  (~2.6 MB)

  rnn_proj_wmma<<<ROWS / 16, 32, 0, stream>>>(x, W_ih, b_ih, b_hh, xw);
  rnn_recurrence<<<1, 32, 0, stream>>>(xw, W_hh, hs);
  rnn_linear<<<(ROWS + 255) / 256, 256, 0, stream>>>(hs, W_lin, b_lin, out);
}